// FinePreprocess_45603962749399
// MI455X (gfx1250) — compile-verified
//
#include <hip/hip_runtime.h>

// CDNA5 / gfx1250, wave32. All matrix math on V_WMMA_F32_16X16X4_F32 (fp32
// matrix pipe) since the problem is gather/memory-bound (~160MB traffic vs
// ~10 GFLOP) and the reference is fp32 end-to-end.

typedef float v2f __attribute__((ext_vector_type(2)));
typedef float v8f __attribute__((ext_vector_type(8)));

#define M_MATCH 3000
#define CF      128
#define HF      240
#define WF      320
#define HW      (HF * WF)     // 76800
#define L_CO    4800
#define WC_GRID 80
#define D_CC    256
#define D_FF    128
#define STRIDEK 4
#define PADW    2

// ---------------------------------------------------------------------------
// Kernel A: W_combT[n, kdc] = sum_j w_down[j, kdc] * w_merge[n, 128 + j]
//   stored transposed ([n][kdc], 128x256) so downstream B-operand loads of
//   (kk, kk+1) at fixed n are one contiguous b64.
// Grid: 128 blocks x 32 threads; block = (ntile 0..7, ktile 0..15), one wave
// computes a 16(n) x 16(kdc) tile with K=128 over j (32 WMMA steps).
// ---------------------------------------------------------------------------
__global__ void wcomb_kernel(const float* __restrict__ w_down,
                             const float* __restrict__ w_merge,
                             float* __restrict__ wcombT) {
  int lane  = threadIdx.x;
  int ntile = blockIdx.x >> 4;
  int ktile = blockIdx.x & 15;
  int half  = lane >> 4;     // which K-pair this lane holds
  int l16   = lane & 15;
  int n  = ntile * 16 + l16;   // A-matrix row (M dim)
  int kd = ktile * 16 + l16;   // B-matrix col (N dim)
  v8f acc = {};
  for (int j = 0; j < D_FF; j += 4) {
    int jj = j + 2 * half;
    v2f av, bv;
    av.x = w_merge[n * (2 * D_FF) + D_FF + jj];      // contiguous b64
    av.y = w_merge[n * (2 * D_FF) + D_FF + jj + 1];
    bv.x = w_down[jj * D_CC + kd];                   // stride-256 pair
    bv.y = w_down[(jj + 1) * D_CC + kd];
    acc = __builtin_amdgcn_wmma_f32_16x16x4_f32(false, av, false, bv,
                                                (short)0, acc, false, false);
  }
#pragma unroll
  for (int v = 0; v < 8; ++v) {
    int nn = ntile * 16 + v + 8 * half;   // D row M = v + 8*(lane/16)
    wcombT[nn * D_CC + kd] = acc[v];
  }
}

// ---------------------------------------------------------------------------
// Kernel B: bias_comb[n] = b_merge[n] + sum_j b_down[j] * w_merge[n, 128+j]
// ---------------------------------------------------------------------------
__global__ void bias_kernel(const float* __restrict__ b_down,
                            const float* __restrict__ b_merge,
                            const float* __restrict__ w_merge,
                            float* __restrict__ bias_comb) {
  int n = threadIdx.x;  // 128 threads
  float s = b_merge[n];
  for (int j = 0; j < D_FF; ++j)
    s += b_down[j] * w_merge[n * (2 * D_FF) + D_FF + j];
  bias_comb[n] = s;
}

// ---------------------------------------------------------------------------
// Kernel C: t[g, :] = c[g, :] @ W_comb + bias_comb   (g in [0, 6000))
//   c[g] gathered from feat_c0/feat_c1 (contiguous 256-float rows, coalesced
//   stage into LDS). One wave per block, 16 g-rows per wave, K=256.
// Grid: 375 blocks x 32 threads (6000/16 exact).
// ---------------------------------------------------------------------------
__global__ void down_kernel(const float* __restrict__ feat_c0,
                            const float* __restrict__ feat_c1,
                            const int* __restrict__ b_ids,
                            const int* __restrict__ i_ids,
                            const int* __restrict__ j_ids,
                            const float* __restrict__ wcombT,
                            const float* __restrict__ bias_comb,
                            float* __restrict__ t_buf) {
  __shared__ float A[16 * D_CC];  // 16 KB
  int lane = threadIdx.x;
  int row0 = blockIdx.x * 16;
  // Stage 16 x 256 coarse rows (consecutive lanes -> consecutive channels).
  for (int it = 0; it < 128; ++it) {
    int idx = it * 32 + lane;
    int r  = idx >> 8;
    int ch = idx & (D_CC - 1);
    int g = row0 + r;
    int mm, id;
    const float* src;
    if (g < M_MATCH) { mm = g;           src = feat_c0; id = i_ids[mm]; }
    else             { mm = g - M_MATCH; src = feat_c1; id = j_ids[mm]; }
    int b = b_ids[mm];
    A[r * D_CC + ch] = src[((size_t)b * L_CO + id) * D_CC + ch];
  }
  __syncthreads();  // single-wave WG: NOP in HW, compiler ordering only
  int half = lane >> 4;
  int l16  = lane & 15;
  for (int nt = 0; nt < 8; ++nt) {
    int n = nt * 16 + l16;
    v8f acc = {};
    for (int k = 0; k < D_CC; k += 4) {
      int kk = k + 2 * half;
      v2f av, bv;
      av.x = A[l16 * D_CC + kk];          // ds b64
      av.y = A[l16 * D_CC + kk + 1];
      bv.x = wcombT[n * D_CC + kk];       // global b64 (hot in cache)
      bv.y = wcombT[n * D_CC + kk + 1];
      acc = __builtin_amdgcn_wmma_f32_16x16x4_f32(false, av, false, bv,
                                                  (short)0, acc, false, false);
    }
    float bias = bias_comb[n];
#pragma unroll
    for (int v = 0; v < 8; ++v) {
      int g = row0 + v + 8 * half;
      t_buf[(size_t)g * D_FF + n] = acc[v] + bias;
    }
  }
}

// ---------------------------------------------------------------------------
// Kernel D: merged[gr, :] = f[gr, :] @ Wf + t[gr/25, :]
//   gr in [0, 150000): gr = m*25 + p, window position p = wr*5 + wc.
//   f[gr, ch] gathered from the padded 5x5 window of feat_f{0,1}.
// Grid: 9375 blocks x 32 threads (150000/16 exact), one wave per block.
// ---------------------------------------------------------------------------
__global__ void merge_kernel(const float* __restrict__ feat_f0,
                             const float* __restrict__ feat_f1,
                             const int* __restrict__ b_ids,
                             const int* __restrict__ i_ids,
                             const int* __restrict__ j_ids,
                             const float* __restrict__ w_merge,
                             const float* __restrict__ t_buf,
                             float* __restrict__ out) {
  __shared__ float A[16 * D_FF];  // 8 KB
  int lane = threadIdx.x;
  int row0 = blockIdx.x * 16;
  // Stage 16 rows x 128 channels of window features (zero-padded OOB).
  for (int it = 0; it < 64; ++it) {
    int idx = it * 32 + lane;
    int r  = idx >> 7;
    int ch = idx & (D_FF - 1);
    int gr = row0 + r;
    int m  = gr / 25;
    int p  = gr - m * 25;
    int wr = p / 5;
    int wc = p - wr * 5;
    int mm, id;
    const float* src;
    if (m < M_MATCH) { mm = m;           src = feat_f0; id = i_ids[mm]; }
    else             { mm = m - M_MATCH; src = feat_f1; id = j_ids[mm]; }
    int b  = b_ids[mm];
    int rr = id / WC_GRID;
    int cc = id - rr * WC_GRID;
    int y = rr * STRIDEK + wr - PADW;
    int x = cc * STRIDEK + wc - PADW;
    float v = 0.0f;
    if (y >= 0 && y < HF && x >= 0 && x < WF)
      v = src[(size_t)b * CF * HW + (size_t)ch * HW + y * WF + x];
    A[r * D_FF + ch] = v;
  }
  __syncthreads();
  int half = lane >> 4;
  int l16  = lane & 15;
  for (int nt = 0; nt < 8; ++nt) {
    int n = nt * 16 + l16;
    v8f acc = {};
    for (int k = 0; k < D_FF; k += 4) {
      int kk = k + 2 * half;
      v2f av, bv;
      av.x = A[l16 * D_FF + kk];                 // ds b64
      av.y = A[l16 * D_FF + kk + 1];
      bv.x = w_merge[n * (2 * D_FF) + kk];       // Wf half, contiguous b64
      bv.y = w_merge[n * (2 * D_FF) + kk + 1];
      acc = __builtin_amdgcn_wmma_f32_16x16x4_f32(false, av, false, bv,
                                                  (short)0, acc, false, false);
    }
#pragma unroll
    for (int v = 0; v < 8; ++v) {
      int gr = row0 + v + 8 * half;
      int m  = gr / 25;
      out[(size_t)gr * D_FF + n] = acc[v] + t_buf[(size_t)m * D_FF + n];
    }
  }
}

// ---------------------------------------------------------------------------
extern "C" void kernel_launch(void* const* d_in, const int* in_sizes, int n_in,
                              void* d_out, int out_size, void* d_ws, size_t ws_size,
                              hipStream_t stream) {
  const float* feat_f0 = (const float*)d_in[0];
  const float* feat_f1 = (const float*)d_in[1];
  const float* feat_c0 = (const float*)d_in[2];
  const float* feat_c1 = (const float*)d_in[3];
  const float* w_down  = (const float*)d_in[4];
  const float* b_down  = (const float*)d_in[5];
  const float* w_merge = (const float*)d_in[6];
  const float* b_merge = (const float*)d_in[7];
  const int*   b_ids   = (const int*)d_in[8];
  const int*   i_ids   = (const int*)d_in[9];
  const int*   j_ids   = (const int*)d_in[10];
  // d_in[11] = stride (==4, hardcoded)

  float* out = (float*)d_out;
  float* ws  = (float*)d_ws;
  float* t_buf     = ws;                       // 6000*128   = 768000 floats
  float* wcombT    = ws + 768000;              // 128*256    =  32768 floats
  float* bias_comb = ws + 768000 + 32768;      // 128 floats  (~3.2 MB total)

  wcomb_kernel<<<128, 32, 0, stream>>>(w_down, w_merge, wcombT);
  bias_kernel<<<1, 128, 0, stream>>>(b_down, b_merge, w_merge, bias_comb);
  down_kernel<<<375, 32, 0, stream>>>(feat_c0, feat_c1, b_ids, i_ids, j_ids,
                                      wcombT, bias_comb, t_buf);
  merge_kernel<<<9375, 32, 0, stream>>>(feat_f0, feat_f1, b_ids, i_ids, j_ids,
                                        w_merge, t_buf, out);
}